// AR_32057635897669
// MI455X (gfx1250) — compile-verified
//
#include <hip/hip_runtime.h>
#include <math.h>

// Problem constants (from reference)
#define M_DIM 4096
#define T_DIM 2048
#define WIN   25
#define OFF   24            // WINDOW - 1
#define N_OUT (T_DIM - OFF) // 2024
#define K_DIM M_DIM         // GEMM inner dim

typedef __attribute__((ext_vector_type(2))) float v2f;
typedef __attribute__((ext_vector_type(8))) float v8f;

__device__ __forceinline__ float softplus_f(float x) {
    // numerically stable: max(x,0) + log1p(exp(-|x|))
    return fmaxf(x, 0.0f) + log1pf(__expf(-fabsf(x)));
}
__device__ __forceinline__ float sigmoid_f(float x) {
    return 1.0f / (1.0f + __expf(-x));
}

// ---------------------------------------------------------------------------
// Kernel 1: softplus(depthwise-corr(ys, repro)) -> d_out  (accumulation seed)
// One block = one (row, 256-output-column) chunk.
// ---------------------------------------------------------------------------
__global__ __launch_bounds__(256) void conv_softplus_kernel(
        const float* __restrict__ ys,
        const float* __restrict__ repro,
        float* __restrict__ out) {
    __shared__ float s_ys[256 + OFF];
    __shared__ float s_rep[WIN];

    const int row = blockIdx.y;
    const int n0  = blockIdx.x * 256;
    const int tid = threadIdx.x;

    for (int i = tid; i < 256 + OFF; i += 256) {
        const int c = n0 + i;
        s_ys[i] = (c < T_DIM) ? ys[(size_t)row * T_DIM + c] : 0.0f;
    }
    if (tid < WIN) s_rep[tid] = repro[row * WIN + tid];
    __syncthreads();

    const int n = n0 + tid;
    if (n < N_OUT) {
        float z = 0.0f;
        #pragma unroll
        for (int k = 0; k < WIN; ++k)
            z = fmaf(s_ys[tid + k], s_rep[k], z);
        out[(size_t)row * N_OUT + n] = softplus_f(z);
    }
}

// ---------------------------------------------------------------------------
// Kernel 2: WMMA fp32 GEMM  G = sigmoid(alphas, diag=0) @ ys[:,24:]
// then out = beta * (out_prev + G)     (out_prev = softplus(Z) from kernel 1)
//
// Block: 256 threads = 8 wave32s. Block tile 128(M) x 256(N), BK = 16.
// Wave grid 2(M) x 4(N); wave tile 64 x 64 -> 4x4 fragments of 16x16
// (128 accumulator VGPRs/lane; 16 WMMA per 8 LDS b64 fragment loads).
//
// Double-buffered pipeline: global loads for K-tile i+1 are issued before
// the barrier so their latency hides under tile i's 64 WMMAs; LDS ping-pongs
// (one barrier/iteration: iteration i+1's barrier separates compute(i) on
// buffer b from the next store into buffer b at iteration i+2).
//
// LDS: A as [m][k], B transposed as [n][k], padded even stride 18 ->
// 8B-aligned ds_load_b64 fragment reads, conflict-free (18r distinct mod 64).
// V_WMMA_F32_16X16X4_F32 fragment layout (wave32):
//   A (16x4): lane L -> M=L%16, holds {k,k+1} (L<16) or {k+2,k+3} (L>=16)
//   B (4x16): lane L -> N=L%16, same K split  -> both are one b64 from [x][k]
//   C/D (16x16, v8f): vgpr r, lane L -> M = r + 8*(L/16), N = L%16
// ---------------------------------------------------------------------------
#define BM 128
#define BN 256
#define BK 16
#define LDK 18       // padded k-stride (even => 8B-aligned b64 LDS loads)
#define NKT (K_DIM / BK)

__global__ __launch_bounds__(256) void gemm_epilogue_kernel(
        const float* __restrict__ ys,
        const float* __restrict__ alphas,
        const float* __restrict__ b0,
        const float* __restrict__ b1,
        float* __restrict__ out) {
    __shared__ float As[2][BM * LDK];  // [m][k]
    __shared__ float Bs[2][BN * LDK];  // [n][k]  (transposed)

    const int tid   = threadIdx.x;
    const int lane  = tid & 31;
    const int wid   = tid >> 5;      // 0..7
    const int waveM = wid & 1;       // 2 waves along M
    const int waveN = wid >> 1;      // 4 waves along N
    const int bm0   = blockIdx.y * BM;
    const int bn0   = blockIdx.x * BN;

    const int nl  = lane & 15;       // fragment row (A) / col (B) / col (C)
    const int mh  = lane >> 4;       // lane-half select
    const int klo = mh << 1;         // 0 or 2: K pair within a K=4 slice

    v8f acc[4][4];
    #pragma unroll
    for (int mi = 0; mi < 4; ++mi)
        #pragma unroll
        for (int ni = 0; ni < 4; ++ni)
            acc[mi][ni] = (v8f){0.f,0.f,0.f,0.f,0.f,0.f,0.f,0.f};

    // ---- staging-register mappings (float4 granular) -----------------------
    // A tile: 128 x 16 = 512 float4 -> 2/thread:  row = idx/4, kc = (idx%4)*4
    // B tile: 16 x 256 = 1024 float4 -> 4/thread: k = idx/64, n = (idx%64)*4
    float4 ra[2], rb[4];

    // prologue: load K-tile 0 into registers
    #pragma unroll
    for (int it = 0; it < 2; ++it) {
        const int idx = tid + it * 256;
        ra[it] = *(const float4*)(alphas
                 + (size_t)(bm0 + (idx >> 2)) * K_DIM + ((idx & 3) << 2));
    }
    #pragma unroll
    for (int it = 0; it < 4; ++it) {
        const int idx = tid + it * 256;
        const int n = (idx & 63) << 2;
        rb[it] = make_float4(0.f, 0.f, 0.f, 0.f);
        if (bn0 + n < N_OUT)   // N_OUT % 4 == 0: float4 chunks all-or-none
            rb[it] = *(const float4*)(ys + (size_t)(idx >> 6) * T_DIM
                                      + OFF + bn0 + n);
    }

    for (int i = 0; i < NKT; ++i) {
        float* __restrict__ Ad = As[i & 1];
        float* __restrict__ Bd = Bs[i & 1];

        // ---- commit staged registers to LDS (A: sigmoid + diag mask) ------
        #pragma unroll
        for (int it = 0; it < 2; ++it) {
            const int idx = tid + it * 256;
            const int row = idx >> 2;
            const int kc  = (idx & 3) << 2;
            const int gm  = bm0 + row;
            const int gk  = i * BK + kc;
            float* dst = &Ad[row * LDK + kc];
            dst[0] = (gm == gk + 0) ? 0.0f : sigmoid_f(ra[it].x);
            dst[1] = (gm == gk + 1) ? 0.0f : sigmoid_f(ra[it].y);
            dst[2] = (gm == gk + 2) ? 0.0f : sigmoid_f(ra[it].z);
            dst[3] = (gm == gk + 3) ? 0.0f : sigmoid_f(ra[it].w);
        }
        #pragma unroll
        for (int it = 0; it < 4; ++it) {
            const int idx = tid + it * 256;
            const int k = idx >> 6;
            const int n = (idx & 63) << 2;
            Bd[(n + 0) * LDK + k] = rb[it].x;
            Bd[(n + 1) * LDK + k] = rb[it].y;
            Bd[(n + 2) * LDK + k] = rb[it].z;
            Bd[(n + 3) * LDK + k] = rb[it].w;
        }

        // ---- issue global loads for K-tile i+1 (latency hidden by WMMAs) --
        if (i + 1 < NKT) {
            const int kt = (i + 1) * BK;
            #pragma unroll
            for (int it = 0; it < 2; ++it) {
                const int idx = tid + it * 256;
                ra[it] = *(const float4*)(alphas
                         + (size_t)(bm0 + (idx >> 2)) * K_DIM
                         + kt + ((idx & 3) << 2));
            }
            #pragma unroll
            for (int it = 0; it < 4; ++it) {
                const int idx = tid + it * 256;
                const int n = (idx & 63) << 2;
                rb[it] = make_float4(0.f, 0.f, 0.f, 0.f);
                if (bn0 + n < N_OUT)
                    rb[it] = *(const float4*)(ys
                             + (size_t)(kt + (idx >> 6)) * T_DIM
                             + OFF + bn0 + n);
            }
        }
        __syncthreads();   // staged LDS visible; also fences buffer reuse

        // ---- compute: 4 K-slices of 4, 16 WMMAs each ----------------------
        #pragma unroll
        for (int kk = 0; kk < BK; kk += 4) {
            v2f af[4], bf[4];
            #pragma unroll
            for (int mi = 0; mi < 4; ++mi)
                af[mi] = *(const v2f*)&Ad[(waveM * 64 + mi * 16 + nl) * LDK
                                          + kk + klo];
            #pragma unroll
            for (int ni = 0; ni < 4; ++ni)
                bf[ni] = *(const v2f*)&Bd[(waveN * 64 + ni * 16 + nl) * LDK
                                          + kk + klo];
            #pragma unroll
            for (int mi = 0; mi < 4; ++mi)
                #pragma unroll
                for (int ni = 0; ni < 4; ++ni)
                    acc[mi][ni] = __builtin_amdgcn_wmma_f32_16x16x4_f32(
                        /*neg_a=*/false, af[mi],
                        /*neg_b=*/false, bf[ni],
                        /*c_mod=*/(short)0, acc[mi][ni],
                        /*reuse_a=*/false, /*reuse_b=*/false);
        }
    }

    // ---- epilogue: out = softplus(b0 + b1*t) * (softplus(Z) + G) ----------
    #pragma unroll
    for (int mi = 0; mi < 4; ++mi) {
        #pragma unroll
        for (int r = 0; r < 8; ++r) {
            const int gm = bm0 + waveM * 64 + mi * 16 + mh * 8 + r;
            const float c0 = b0[gm];
            const float c1 = b1[gm];
            #pragma unroll
            for (int ni = 0; ni < 4; ++ni) {
                const int gn = bn0 + waveN * 64 + ni * 16 + nl;
                if (gn < N_OUT) {
                    const float beta =
                        softplus_f(fmaf(c1, (float)(gn + WIN), c0));
                    const size_t o = (size_t)gm * N_OUT + gn;
                    out[o] = beta * (out[o] + acc[mi][ni][r]);
                }
            }
        }
    }
}

// ---------------------------------------------------------------------------
extern "C" void kernel_launch(void* const* d_in, const int* in_sizes, int n_in,
                              void* d_out, int out_size, void* d_ws,
                              size_t ws_size, hipStream_t stream) {
    const float* ys     = (const float*)d_in[0];
    const float* alphas = (const float*)d_in[1];
    const float* repro  = (const float*)d_in[2];
    const float* b0     = (const float*)d_in[3];
    const float* b1     = (const float*)d_in[4];
    // d_in[5] is t = arange(T)+1; values are implied by the column index.
    float* out = (float*)d_out;

    // 1) seed d_out with softplus(conv(ys, repro))
    dim3 g1((N_OUT + 255) / 256, M_DIM);
    conv_softplus_kernel<<<g1, 256, 0, stream>>>(ys, repro, out);

    // 2) WMMA GEMM + fused beta epilogue (reads + overwrites d_out)
    dim3 g2((N_OUT + BN - 1) / BN, M_DIM / BM);
    gemm_epilogue_kernel<<<g2, 256, 0, stream>>>(ys, alphas, b0, b1, out);
}